// EdgeUpdate_16982300688556
// MI455X (gfx1250) — compile-verified
//
#include <hip/hip_runtime.h>
#include <math.h>

typedef float v2f __attribute__((ext_vector_type(2)));
typedef float v8f __attribute__((ext_vector_type(8)));

#define IN_DIM   192
#define HID      128
#define OUT_DIM  64
#define NODE_F   64
#define EDGE_F   64

#define EPW      64      // edges per wave (4 M-tiles of 16)
#define MT       4
#define CPAD     196     // combined row stride in words; 196%64==4 -> conflict-free A fetch
#define HPAD     132     // h row stride in words;       132%64==4 -> conflict-free A fetch

#define W1P_FLOATS (8*48*64)   // 24576 floats = 96 KB (packed W1, WMMA-B layout)
#define W2P_FLOATS (4*32*64)   // 8192  floats = 32 KB (packed W2, WMMA-B layout)

// Repack W1 (128x192) and W2 (64x128) into the native WMMA B-operand layout:
// for tile (nt, kq): lane l supplies B(k,n) with n = nt*16 + (l&15),
// k = 4*kq + 2*(l>>4) + j (j = vgpr 0/1). Stored so lane l's data is a
// contiguous b64 at base + l*8  -> fully coalesced 256B per fetch.
__global__ __launch_bounds__(64) void pack_weights(const float* __restrict__ W1,
                                                   const float* __restrict__ W2,
                                                   float* __restrict__ w1p,
                                                   float* __restrict__ w2p)
{
    int t = blockIdx.x * blockDim.x + threadIdx.x;
    int stride = gridDim.x * blockDim.x;
    for (int i = t; i < W1P_FLOATS; i += stride) {
        int j  = i & 1;
        int l  = (i >> 1) & 31;
        int kq = (i >> 6) % 48;
        int nt = (i >> 6) / 48;
        int n = nt * 16 + (l & 15);
        int k = 4 * kq + 2 * (l >> 4) + j;
        w1p[i] = W1[n * IN_DIM + k];
    }
    for (int i = t; i < W2P_FLOATS; i += stride) {
        int j  = i & 1;
        int l  = (i >> 1) & 31;
        int kq = (i >> 6) & 31;
        int nt = (i >> 6) >> 5;
        int n = nt * 16 + (l & 15);
        int k = 4 * kq + 2 * (l >> 4) + j;
        w2p[i] = W2[n * HID + k];
    }
}

// stable shifted-softplus on the hardware TRANS pipe:
// softplus(x) - ln2 = max(x,0) + log(1 + exp(-|x|)) - ln2
// __expf/__logf lower to v_exp_f32/v_log_f32 (~2 ulp); the log term is
// bounded by ln2 so absolute error stays ~1e-7 on an O(1) output.
__device__ __forceinline__ float softplus_shift(float x)
{
    const float LN2 = 0.6931471805599453f;
    return fmaxf(x, 0.0f) + __logf(1.0f + __expf(-fabsf(x))) - LN2;
}

__global__ __launch_bounds__(32) void edge_mlp(const float* __restrict__ edge_state,
                                               const int*   __restrict__ edges,
                                               const float* __restrict__ node_state,
                                               const float* __restrict__ w1p,
                                               const float* __restrict__ b1,
                                               const float* __restrict__ w2p,
                                               const float* __restrict__ b2,
                                               float* __restrict__ out,
                                               int E)
{
    __shared__ float s_comb[EPW * CPAD];  // 50176 B
    __shared__ float s_h[EPW * HPAD];     // 33792 B

    const int lane = threadIdx.x;         // 0..31 (one full wave32)
    const int nlo  = lane & 15;
    const int khi  = (lane >> 4) * 2;
    const int base = blockIdx.x * EPW;

    // ---- Phase A: gather + stage combined = [node[i0] | node[i1] | edge] ----
    for (int e = 0; e < EPW; ++e) {
        int ge = base + e;
        if (ge >= E) ge = E - 1;                 // keep EXEC full; dup rows, stores guarded
        int i0 = edges[2 * ge + 0];              // uniform -> s_load
        int i1 = edges[2 * ge + 1];
        v2f a = *(const v2f*)(node_state + (size_t)i0 * NODE_F + 2 * lane);
        v2f b = *(const v2f*)(node_state + (size_t)i1 * NODE_F + 2 * lane);
        v2f c = *(const v2f*)(edge_state + (size_t)ge * EDGE_F + 2 * lane);
        *(v2f*)&s_comb[e * CPAD +       2 * lane] = a;
        *(v2f*)&s_comb[e * CPAD +  64 + 2 * lane] = b;
        *(v2f*)&s_comb[e * CPAD + 128 + 2 * lane] = c;
    }
    __syncthreads();

    // ---- Layer 1: h = softplus(combined @ W1^T + b1) - ln2 ----
    for (int nt = 0; nt < 8; ++nt) {
        float bias = b1[nt * 16 + nlo];
        v8f acc[MT];
#pragma unroll
        for (int mt = 0; mt < MT; ++mt)
            acc[mt] = (v8f){bias, bias, bias, bias, bias, bias, bias, bias};

        const float* wb = w1p + nt * (48 * 64);
#pragma unroll
        for (int kq = 0; kq < 48; ++kq) {
            v2f bv = *(const v2f*)(wb + kq * 64 + 2 * lane);   // coalesced 256B from WGP$
            int ko = 4 * kq + khi;
#pragma unroll
            for (int mt = 0; mt < MT; ++mt) {
                v2f av = *(const v2f*)&s_comb[(mt * 16 + nlo) * CPAD + ko];
                acc[mt] = __builtin_amdgcn_wmma_f32_16x16x4_f32(
                    false, av, false, bv, (short)0, acc[mt], false, false);
            }
        }
        // activation + store h tile (D layout: vgpr r -> M=r+8*(lane/16), N=lane&15)
#pragma unroll
        for (int mt = 0; mt < MT; ++mt) {
#pragma unroll
            for (int r = 0; r < 8; ++r) {
                float sp = softplus_shift(acc[mt][r]);
                int m = mt * 16 + r + (lane >> 4) * 8;
                s_h[m * HPAD + nt * 16 + nlo] = sp;
            }
        }
    }
    __syncthreads();

    // ---- Layer 2: out = h @ W2^T + b2 ----
    for (int nt = 0; nt < 4; ++nt) {
        float bias = b2[nt * 16 + nlo];
        v8f acc[MT];
#pragma unroll
        for (int mt = 0; mt < MT; ++mt)
            acc[mt] = (v8f){bias, bias, bias, bias, bias, bias, bias, bias};

        const float* wb = w2p + nt * (32 * 64);
#pragma unroll
        for (int kq = 0; kq < 32; ++kq) {
            v2f bv = *(const v2f*)(wb + kq * 64 + 2 * lane);
            int ko = 4 * kq + khi;
#pragma unroll
            for (int mt = 0; mt < MT; ++mt) {
                v2f av = *(const v2f*)&s_h[(mt * 16 + nlo) * HPAD + ko];
                acc[mt] = __builtin_amdgcn_wmma_f32_16x16x4_f32(
                    false, av, false, bv, (short)0, acc[mt], false, false);
            }
        }
#pragma unroll
        for (int mt = 0; mt < MT; ++mt) {
#pragma unroll
            for (int r = 0; r < 8; ++r) {
                int m = base + mt * 16 + r + (lane >> 4) * 8;
                if (m < E)
                    out[(size_t)m * OUT_DIM + nt * 16 + nlo] = acc[mt][r];
            }
        }
    }
}

extern "C" void kernel_launch(void* const* d_in, const int* in_sizes, int n_in,
                              void* d_out, int out_size, void* d_ws, size_t ws_size,
                              hipStream_t stream) {
    const float* edge_state = (const float*)d_in[0];
    const int*   edges      = (const int*)d_in[1];
    const float* node_state = (const float*)d_in[2];
    const float* W1         = (const float*)d_in[3];
    const float* b1         = (const float*)d_in[4];
    const float* W2         = (const float*)d_in[5];
    const float* b2         = (const float*)d_in[6];
    float* out = (float*)d_out;

    int E = in_sizes[0] / EDGE_F;

    float* w1p = (float*)d_ws;             // 96 KB
    float* w2p = w1p + W1P_FLOATS;         // 32 KB  (total 128 KB of d_ws)

    pack_weights<<<64, 64, 0, stream>>>(W1, W2, w1p, w2p);

    int blocks = (E + EPW - 1) / EPW;
    edge_mlp<<<blocks, 32, 0, stream>>>(edge_state, edges, node_state,
                                        w1p, b1, w2p, b2, out, E);
}